// BatchRelationalEncoder_67044439491169
// MI455X (gfx1250) — compile-verified
//
#include <hip/hip_runtime.h>

#define C 128
#define NBASES 4
#define SA_STRIDE 132   // 64-row A slab, padded: bank = 4*lr + c -> conflict-free
#define SB_STRIDE 48    // 128-row B tile, padded: 4 loads/k-step hit disjoint bank groups

typedef __attribute__((ext_vector_type(2))) float v2f;
typedef __attribute__((ext_vector_type(8))) float v8f;

__device__ __forceinline__ float wave_sum32(float v) {
    #pragma unroll
    for (int off = 16; off > 0; off >>= 1)
        v += __shfl_xor(v, off, 32);
    return v;
}

__global__ void zero_f(float* __restrict__ p, int n) {
    int i = blockIdx.x * blockDim.x + threadIdx.x;
    if (i < n) p[i] = 0.0f;
}

__global__ void degree_k(const int* __restrict__ tri, float* __restrict__ deg, int nE) {
    int e = blockIdx.x * blockDim.x + threadIdx.x;
    if (e < nE) unsafeAtomicAdd(&deg[tri[e * 3 + 2]], 1.0f);
}

__global__ void deginv_k(float* __restrict__ deg, int n) {
    int i = blockIdx.x * blockDim.x + threadIdx.x;
    if (i < n) {
        float d = deg[i];
        deg[i] = (d > 0.0f) ? (1.0f / d) : 0.0f;
    }
}

// Wcat (128 x 640 row-major) = [basis0 | basis1 | basis2 | basis3 | root_w]
__global__ void build_wcat(const float* __restrict__ basis, const float* __restrict__ rootw,
                           float* __restrict__ w) {
    int idx = blockIdx.x * blockDim.x + threadIdx.x;
    if (idx >= 128 * 640) return;
    int i = idx / 640, j = idx % 640;
    float v = (j < 512) ? basis[(j >> 7) * C * C + i * C + (j & 127)]
                        : rootw[i * C + (j - 512)];
    w[idx] = v;
}

// One wave computes a 16x32 fp32 tile (two 16x16 WMMA accs sharing the A fragment)
// via V_WMMA_F32_16X16X4_F32 over K=128. Block = 128 threads (4 waves) = 64 A rows.
// Columns j < split -> Y0 as xb[j>>7][row][j&127] (+bias0); else Y1[row][j-split] (+bias1).
__global__ __launch_bounds__(128)
void gemm_wmma(const float* __restrict__ A, const float* __restrict__ W,
               float* __restrict__ Y0, const float* __restrict__ bias0,
               float* __restrict__ Y1, const float* __restrict__ bias1,
               int nrows, int ncols, int split) {
    __shared__ float sA[64 * SA_STRIDE];    // ~33.8 KB
    __shared__ float sB[128 * SB_STRIDE];   // 24 KB
    const int t = threadIdx.x;
    const int wave = t >> 5, lane = t & 31;
    const int half = lane >> 4, lr = lane & 15;
    const int rowBase = blockIdx.x * 64;
    const int n0 = blockIdx.y * 32;

    // Stage A slab (64x128): 2048 float4, 16 per thread, coalesced.
    const float4* A4 = reinterpret_cast<const float4*>(A);
    #pragma unroll
    for (int i = 0; i < 16; ++i) {
        int idx4 = t + i * 128;
        int r = idx4 >> 5, c4 = idx4 & 31;   // 32 float4 per logical row
        int gr = rowBase + r;
        float4 v = make_float4(0.f, 0.f, 0.f, 0.f);
        if (gr < nrows) v = A4[(size_t)gr * 32 + c4];
        *reinterpret_cast<float4*>(&sA[r * SA_STRIDE + c4 * 4]) = v;  // 528B row pitch, 16B aligned
    }
    // Stage B tile (128x32): 4096 floats, 32 per thread.
    #pragma unroll
    for (int i = 0; i < 32; ++i) {
        int e = t + i * 128;
        int r = e >> 5, cc = e & 31;
        sB[r * SB_STRIDE + cc] = W[(size_t)r * ncols + n0 + cc];
    }
    __syncthreads();

    // A 16x4 layout: lane l (M = l&15) holds K = v + 2*(l>>4) in VGPR v.
    // B 4x16 layout: VGPR v, lanes 0-15: K = v + 2*half, N = lane&15 (tile0) / +16 (tile1).
    v8f acc0 = {0.f, 0.f, 0.f, 0.f, 0.f, 0.f, 0.f, 0.f};
    v8f acc1 = acc0;
    const int m0l = wave * 16;
    const float* pa = &sA[(m0l + lr) * SA_STRIDE + 2 * half];
    const float* pb = &sB[(2 * half) * SB_STRIDE + lr];
    #pragma unroll 8
    for (int k = 0; k < 128; k += 4) {
        v2f a, b0, b1;
        a.x  = pa[k];
        a.y  = pa[k + 1];
        b0.x = pb[(size_t)k * SB_STRIDE];
        b0.y = pb[(size_t)(k + 1) * SB_STRIDE];
        b1.x = pb[(size_t)k * SB_STRIDE + 16];
        b1.y = pb[(size_t)(k + 1) * SB_STRIDE + 16];
        acc0 = __builtin_amdgcn_wmma_f32_16x16x4_f32(
            false, a, false, b0, (short)0, acc0, false, false);
        acc1 = __builtin_amdgcn_wmma_f32_16x16x4_f32(
            false, a, false, b1, (short)0, acc1, false, false);
    }

    // C/D layout: VGPR r, lanes 0-15 -> M = r, lanes 16-31 -> M = r + 8; N = lane&15.
    #pragma unroll
    for (int r = 0; r < 8; ++r) {
        int grow = rowBase + m0l + r + 8 * half;
        if (grow >= nrows) continue;
        #pragma unroll
        for (int s = 0; s < 2; ++s) {
            int j = n0 + s * 16 + lr;
            float v = (s == 0) ? acc0[r] : acc1[r];
            if (j < split) {
                int b = j >> 7, o = j & 127;
                if (bias0) v += bias0[o];
                Y0[((size_t)b * nrows + grow) * C + o] = v;
            } else {
                int o = j - split;
                if (bias1) v += bias1[o];
                Y1[(size_t)grow * C + o] = v;
            }
        }
    }
}

// One wave per edge; lane handles 4 channels (float4). Gathers hit L2 (xb = 102.4MB < 192MB L2).
// Edge triple / att row / deg_inv are wave-uniform -> readfirstlane to enable scalar loads.
__global__ __launch_bounds__(256)
void edge_msg(const int* __restrict__ tri, const float* __restrict__ att,
              const float* __restrict__ xb, const float* __restrict__ dinv,
              float* __restrict__ out, int nrows, int nE) {
    int gw = (int)((blockIdx.x * (size_t)blockDim.x + threadIdx.x) >> 5);
    int lane = threadIdx.x & 31;
    if (gw >= nE) return;
    int base3 = __builtin_amdgcn_readfirstlane(gw * 3);
    int src = __builtin_amdgcn_readfirstlane(tri[base3 + 0]);
    int rel = __builtin_amdgcn_readfirstlane(tri[base3 + 1]);
    int dst = __builtin_amdgcn_readfirstlane(tri[base3 + 2]);
    float c0 = att[rel * NBASES + 0];
    float c1 = att[rel * NBASES + 1];
    float c2 = att[rel * NBASES + 2];
    float c3 = att[rel * NBASES + 3];
    float scale = dinv[dst];
    const float4* xb4 = reinterpret_cast<const float4*>(xb);
    size_t base = (size_t)src * 32 + lane;
    size_t stride = (size_t)nrows * 32;
    float4 v0 = xb4[base];
    float4 v1 = xb4[base + stride];
    float4 v2 = xb4[base + 2 * stride];
    float4 v3 = xb4[base + 3 * stride];
    float4 m;
    m.x = (c0 * v0.x + c1 * v1.x + c2 * v2.x + c3 * v3.x) * scale;
    m.y = (c0 * v0.y + c1 * v1.y + c2 * v2.y + c3 * v3.y) * scale;
    m.z = (c0 * v0.z + c1 * v1.z + c2 * v2.z + c3 * v3.z) * scale;
    m.w = (c0 * v0.w + c1 * v1.w + c2 * v2.w + c3 * v3.w) * scale;
    float* o = out + (size_t)dst * C + lane * 4;
    unsafeAtomicAdd(o + 0, m.x);
    unsafeAtomicAdd(o + 1, m.y);
    unsafeAtomicAdd(o + 2, m.z);
    unsafeAtomicAdd(o + 3, m.w);
}

// LayerNorm (+optional ReLU): one wave32 per row of 128.
__global__ __launch_bounds__(256)
void ln_k(const float* __restrict__ in, float* __restrict__ outp,
          const float* __restrict__ g, const float* __restrict__ b,
          int nrows, int do_relu) {
    int row = (int)((blockIdx.x * (size_t)blockDim.x + threadIdx.x) >> 5);
    int lane = threadIdx.x & 31;
    if (row >= nrows) return;
    const float4* r4 = reinterpret_cast<const float4*>(in + (size_t)row * C);
    float4 v = r4[lane];
    float mu = wave_sum32(v.x + v.y + v.z + v.w) * (1.0f / 128.0f);
    float dx = v.x - mu, dy = v.y - mu, dz = v.z - mu, dw = v.w - mu;
    float var = wave_sum32(dx * dx + dy * dy + dz * dz + dw * dw) * (1.0f / 128.0f);
    float rstd = rsqrtf(var + 1e-5f);
    float4 gg = reinterpret_cast<const float4*>(g)[lane];
    float4 bb = reinterpret_cast<const float4*>(b)[lane];
    float4 o;
    o.x = dx * rstd * gg.x + bb.x;
    o.y = dy * rstd * gg.y + bb.y;
    o.z = dz * rstd * gg.z + bb.z;
    o.w = dw * rstd * gg.w + bb.w;
    if (do_relu) {
        o.x = fmaxf(o.x, 0.f); o.y = fmaxf(o.y, 0.f);
        o.z = fmaxf(o.z, 0.f); o.w = fmaxf(o.w, 0.f);
    }
    reinterpret_cast<float4*>(outp + (size_t)row * C)[lane] = o;
}

extern "C" void kernel_launch(void* const* d_in, const int* in_sizes, int n_in,
                              void* d_out, int out_size, void* d_ws, size_t ws_size,
                              hipStream_t stream) {
    const float* node_features = (const float*)d_in[0];
    const int*   tri           = (const int*)d_in[1];
    const float* input_w       = (const float*)d_in[3];
    const float* input_b       = (const float*)d_in[4];
    const float* basis0 = (const float*)d_in[5];
    const float* att0   = (const float*)d_in[6];
    const float* rootw0 = (const float*)d_in[7];
    const float* rootb0 = (const float*)d_in[8];
    const float* lng0   = (const float*)d_in[9];
    const float* lnb0   = (const float*)d_in[10];
    const float* basis1 = (const float*)d_in[11];
    const float* att1   = (const float*)d_in[12];
    const float* rootw1 = (const float*)d_in[13];
    const float* rootb1 = (const float*)d_in[14];
    const float* lng1   = (const float*)d_in[15];
    const float* lnb1   = (const float*)d_in[16];

    const int Nn = in_sizes[0] / C;     // 50000
    const int Ne = in_sizes[1] / 3;     // 800000

    float* x    = (float*)d_ws;                 // N*C
    float* y    = x + (size_t)Nn * C;           // N*C
    float* xb   = y + (size_t)Nn * C;           // 4*N*C
    float* dinv = xb + (size_t)4 * Nn * C;      // N
    float* wcat = dinv + Nn;                    // 128*640

    // Degrees (edges identical across layers -> once).
    zero_f<<<(Nn + 255) / 256, 256, 0, stream>>>(dinv, Nn);
    degree_k<<<(Ne + 255) / 256, 256, 0, stream>>>(tri, dinv, Ne);
    deginv_k<<<(Nn + 255) / 256, 256, 0, stream>>>(dinv, Nn);

    // x = node_features @ input_w + input_b
    {
        dim3 grid((Nn + 63) / 64, C / 32);
        gemm_wmma<<<grid, 128, 0, stream>>>(node_features, input_w,
                                            x, input_b, nullptr, nullptr,
                                            Nn, C, 1 << 30);
    }

    dim3 gridc((Nn + 63) / 64, 640 / 32);
    dim3 gride((unsigned)(((size_t)Ne * 32 + 255) / 256));
    dim3 gridn((unsigned)(((size_t)Nn * 32 + 255) / 256));

    // ---- layer 0 ----
    build_wcat<<<(128 * 640 + 255) / 256, 256, 0, stream>>>(basis0, rootw0, wcat);
    gemm_wmma<<<gridc, 128, 0, stream>>>(x, wcat, xb, nullptr, y, rootb0, Nn, 640, 512);
    edge_msg<<<gride, 256, 0, stream>>>(tri, att0, xb, dinv, y, Nn, Ne);
    ln_k<<<gridn, 256, 0, stream>>>(y, x, lng0, lnb0, Nn, 1);

    // ---- layer 1 ----
    build_wcat<<<(128 * 640 + 255) / 256, 256, 0, stream>>>(basis1, rootw1, wcat);
    gemm_wmma<<<gridc, 128, 0, stream>>>(x, wcat, xb, nullptr, y, rootb1, Nn, 640, 512);
    edge_msg<<<gride, 256, 0, stream>>>(tri, att1, xb, dinv, y, Nn, Ne);
    ln_k<<<gridn, 256, 0, stream>>>(y, (float*)d_out, lng1, lnb1, Nn, 0);
}